// LaslNet_10995116278526
// MI455X (gfx1250) — compile-verified
//
#include <hip/hip_runtime.h>

typedef __attribute__((ext_vector_type(2))) float v2f;
typedef __attribute__((ext_vector_type(8))) float v8f;

#define DIN   36
#define DOUT  34
#define WSTR  52            // slab row stride in floats (cols 0..51 used, 36..51 zero)
#define ROWS  324           // 3*3*36 : (td,te) x full f extent
#define SLABN (ROWS * WSTR) // 16848 floats per slab

__global__ __launch_bounds__(96)
void conv4d_complex_head_kernel(const float* __restrict__ xr,
                                const float* __restrict__ xi,
                                const float* __restrict__ wr,
                                const float* __restrict__ wi,
                                const float* __restrict__ br,
                                const float* __restrict__ bi,
                                const float* __restrict__ hwv,
                                const float* __restrict__ hbv,
                                float* __restrict__ out) {
    __shared__ float s_xr[SLABN];
    __shared__ float s_xi[SLABN];
    __shared__ float s_w[162];      // [0..80]=w_real[0], [81..161]=w_imag[0]
    __shared__ float s_hw[34];
    __shared__ float s_red[3 * 256];

    const int tid = threadIdx.x;
    int blk = blockIdx.x;
    const int e = blk % DOUT; blk /= DOUT;
    const int d = blk % DOUT; blk /= DOUT;
    const int b = blk;

    // ---- cooperative loads: weights, head weights, full slab (all 3 waves) ----
    for (int i = tid; i < 81; i += 96) { s_w[i] = wr[i]; s_w[81 + i] = wi[i]; }
    for (int i = tid; i < 34; i += 96) { s_hw[i] = hwv[i]; }

    for (int i = tid; i < SLABN; i += 96) {
        const int row = i / WSTR, col = i % WSTR;
        const int ff   = row % DIN;          // global f, 0..35
        const int tdte = row / DIN;          // td*3+te
        const int td = tdte / 3, te = tdte % 3;
        const bool ok = (col < DIN);
        const long g = ((((long)b * DIN + (d + td)) * DIN + (e + te)) * DIN + ff) * DIN
                       + (ok ? col : 0);
        s_xr[i] = ok ? xr[g] : 0.f;
        s_xi[i] = ok ? xi[g] : 0.f;
    }
    __syncthreads();

    const float bias_r = br[0];
    const float bias_i = bi[0];

    const int wavei = tid >> 5;        // 0,1,2 -> f-tile
    const int lane  = tid & 31;
    const int f0    = 16 * wavei;
    const int n     = lane & 15;       // N column / A-row site m
    const int hi    = lane >> 4;       // 0: K={0,1}, 1: K={2,3}
    const int m     = n;

    v8f accr[3] = {{}, {}, {}};        // per w-tile: conv(xr,wr) - conv(xi,wi)
    v8f acci[3] = {{}, {}, {}};        // per w-tile: conv(xr,wi) + conv(xi,wr)

    for (int t = 0; t < 27; ++t) {
        const int wb = 3 * t;          // tap base: t = (td*3+te)*3 + tf
        const float gr0 = s_w[wb],      gr1 = s_w[wb + 1],      gr2 = s_w[wb + 2];
        const float gi0 = s_w[81 + wb], gi1 = s_w[81 + wb + 1], gi2 = s_w[81 + wb + 2];
        const int abase = ((t / 3) * DIN + f0 + (t % 3) + m) * WSTR;   // A row base

#pragma unroll
        for (int kstep = 0; kstep < 5; ++kstep) {
            const int kg = 4 * kstep + 2 * hi;     // this lane's K indices kg, kg+1
            const int d0 = kg - n;                 // Toeplitz diagonal for .x
            const int d1 = d0 + 1;                 // for .y

            v2f Brr, Bii;
            Brr.x = (d0 == 0) ? gr0 : (d0 == 1) ? gr1 : (d0 == 2) ? gr2 : 0.f;
            Brr.y = (d1 == 0) ? gr0 : (d1 == 1) ? gr1 : (d1 == 2) ? gr2 : 0.f;
            Bii.x = (d0 == 0) ? gi0 : (d0 == 1) ? gi1 : (d0 == 2) ? gi2 : 0.f;
            Bii.y = (d1 == 0) ? gi0 : (d1 == 1) ? gi1 : (d1 == 2) ? gi2 : 0.f;
            v2f Bni; Bni.x = -Bii.x; Bni.y = -Bii.y;

#pragma unroll
            for (int wt = 0; wt < 3; ++wt) {
                const int col = 16 * wt + kg;      // <= 51 < WSTR
                const v2f axr = *(const v2f*)&s_xr[abase + col];
                const v2f axi = *(const v2f*)&s_xi[abase + col];
                accr[wt] = __builtin_amdgcn_wmma_f32_16x16x4_f32(false, axr, false, Brr, (short)0, accr[wt], false, false);
                accr[wt] = __builtin_amdgcn_wmma_f32_16x16x4_f32(false, axi, false, Bni, (short)0, accr[wt], false, false);
                acci[wt] = __builtin_amdgcn_wmma_f32_16x16x4_f32(false, axr, false, Bii, (short)0, acci[wt], false, false);
                acci[wt] = __builtin_amdgcn_wmma_f32_16x16x4_f32(false, axi, false, Brr, (short)0, acci[wt], false, false);
            }
        }
    }

    // ---- bias + CReLU + modulus + head weight, per (site m, w = 16*wt + n) ----
    float hp[8];
#pragma unroll
    for (int r = 0; r < 8; ++r) hp[r] = 0.f;
#pragma unroll
    for (int wt = 0; wt < 3; ++wt) {
        const int wglob = 16 * wt + n;
        const float hwf = (wglob < DOUT) ? s_hw[wglob] : 0.f;
#pragma unroll
        for (int r = 0; r < 8; ++r) {
            const float yr = fmaxf(accr[wt][r] + bias_r, 0.f);
            const float yi = fmaxf(acci[wt][r] + bias_i, 0.f);
            hp[r] += __builtin_sqrtf(yr * yr + yi * yi + 1e-12f) * hwf;
        }
    }

    // ---- cross-lane reduce over n (16 w columns) per site m, per wave ----
    __syncthreads();
#pragma unroll
    for (int r = 0; r < 8; ++r) {
        const int mm = r + 8 * hi;     // C/D layout: lane<16 -> M=r, else M=r+8
        s_red[wavei * 256 + mm * 16 + n] = hp[r];
    }
    __syncthreads();
    if (lane < 16) {
        float h = 0.f;
#pragma unroll
        for (int j = 0; j < 16; ++j) h += s_red[wavei * 256 + lane * 16 + j];
        h += hbv[0];
        const float o = 1.f / (1.f + __expf(-h));
        const int f = f0 + lane;
        if (f < DOUT)
            out[(((long)b * DOUT + d) * DOUT + e) * DOUT + f] = o;
    }
}

extern "C" void kernel_launch(void* const* d_in, const int* in_sizes, int n_in,
                              void* d_out, int out_size, void* d_ws, size_t ws_size,
                              hipStream_t stream) {
    const float* xr = (const float*)d_in[0];
    const float* xi = (const float*)d_in[1];
    const float* wr = (const float*)d_in[2];   // channel 0 = first 81 floats
    const float* wi = (const float*)d_in[3];
    const float* br = (const float*)d_in[4];
    const float* bi = (const float*)d_in[5];
    const float* hw = (const float*)d_in[6];
    const float* hb = (const float*)d_in[7];
    float* out = (float*)d_out;

    const int nblocks = 8 * DOUT * DOUT;   // b * d * e = 9248; 3 waves cover f-tiles
    conv4d_complex_head_kernel<<<dim3(nblocks), dim3(96), 0, stream>>>(
        xr, xi, wr, wi, br, bi, hw, hb, out);
}